// EdgeLoss_15307263443335
// MI455X (gfx1250) — compile-verified
//
#include <hip/hip_runtime.h>
#include <hip/hip_bf16.h>

#define IMG_W 1024
#define IMG_H 1024
#define N_IMG 32
#define TW 128
#define TH 32
#define LDSW 136                 // (TW + 8) floats per LDS row (16B-aligned segs)
#define LDSH 34                  // TH + 2 halo rows
#define NSEG (LDSH * (LDSW / 4)) // 34*34 = 1156 float4 segments per array
#define NBLK (N_IMG * (IMG_H / TH) * (IMG_W / TW)) // 8192 blocks

typedef float v2f_t __attribute__((ext_vector_type(2)));
typedef float v8f_t __attribute__((ext_vector_type(8)));
typedef int   v4i_t __attribute__((vector_size(16)));   // matches builtin param type

// ---- CDNA5 async global->LDS (ASYNCcnt path, CDNA5 ISA) ----
// Builtin signature (from hipcc diagnostic): (global v4i* src, local v4i* dst,
// imm int offset, imm int cpol)
__device__ __forceinline__ void async_ld_b128(const float* g, float* l) {
#if __has_builtin(__builtin_amdgcn_global_load_async_to_lds_b128)
  __builtin_amdgcn_global_load_async_to_lds_b128(
      (__attribute__((address_space(1))) v4i_t*)g,
      (__attribute__((address_space(3))) v4i_t*)l, 0, 0);
#else
  unsigned loff = (unsigned)(size_t)(__attribute__((address_space(3))) float*)l;
  asm volatile("global_load_async_to_lds_b128 %0, %1, off"
               :: "v"(loff), "v"(g) : "memory");
#endif
}

__device__ __forceinline__ void wait_async0() {
#if __has_builtin(__builtin_amdgcn_s_wait_asynccnt)
  __builtin_amdgcn_s_wait_asynccnt(0);
#else
  asm volatile("s_wait_asynccnt 0x0" ::: "memory");
#endif
}

// ---- wave-0 sum finisher via V_WMMA_F32_16X16X4_F32 ----
// Every lane of wave 0 holds s. A[16x4]: lane L<16 -> A[L][0]=s (v0), A[L][1]=0 (v1);
// lane L>=16 -> A[L-16][2]=s, A[L-16][3]=0. B = ones(4x16) => D[i][j] = rowsum_i(A).
// Sum of the 8 D VGPRs in a lane < 16 = sum of rows 0..7; in a lane >= 16 = rows 8..15.
// Total = t@lane0 + t@lane16. Layout-correct per CDNA5 ISA §7.12.2.
__device__ __forceinline__ float wave_sum_wmma(float s) {
#if __has_builtin(__builtin_amdgcn_wmma_f32_16x16x4_f32)
  v2f_t a; a.x = s;    a.y = 0.0f;
  v2f_t b; b.x = 1.0f; b.y = 1.0f;
  v8f_t c = {};
  v8f_t d = __builtin_amdgcn_wmma_f32_16x16x4_f32(false, a, false, b,
                                                  (short)0, c, false, false);
  float t = d[0] + d[1] + d[2] + d[3] + d[4] + d[5] + d[6] + d[7];
  return __shfl(t, 0, 32) + __shfl(t, 16, 32);
#else
  #pragma unroll
  for (int off = 16; off > 0; off >>= 1) s += __shfl_down(s, off, 32);
  return __shfl(s, 0, 32);
#endif
}

__global__ __launch_bounds__(256)
void edge_sobel_tile(const float* __restrict__ sr, const float* __restrict__ hr,
                     float* __restrict__ partial) {
  __shared__ __align__(16) float sS[LDSH * LDSW];
  __shared__ __align__(16) float sH[LDSH * LDSW];
  __shared__ float red[256];

  const int tid = threadIdx.x;
  const int x0  = blockIdx.x * TW;
  const int y0  = blockIdx.y * TH;
  const size_t base = (size_t)blockIdx.z << 20; // img * 1024*1024

  // ---- Stage sr & hr tiles (with halo + zero padding) into LDS via async DMA ----
  for (int s = tid; s < NSEG; s += 256) {
    const int row  = s / 34;
    const int ch   = s - row * 34;
    const int gy   = y0 - 1 + row;
    const int gx   = x0 - 4 + ch * 4;         // 16B aligned; never straddles edges
    const int loff = row * LDSW + ch * 4;
    if ((unsigned)gy < IMG_H && (unsigned)gx < IMG_W) {
      const size_t gidx = base + (size_t)gy * IMG_W + (size_t)gx;
      async_ld_b128(sr + gidx, &sS[loff]);
      async_ld_b128(hr + gidx, &sH[loff]);
    } else {
      float4 z = make_float4(0.f, 0.f, 0.f, 0.f);
      *(float4*)&sS[loff] = z;
      *(float4*)&sH[loff] = z;
    }
  }
  wait_async0();
  __syncthreads();

  // ---- Fused Sobel on d = sr - hr, 4x4 outputs/thread, rolling 3-row window ----
  const int tx = tid & 31;
  const int ty = tid >> 5;
  const int cb = 4 * tx + 3;   // LDS col of (x-1) for this thread's 4-wide strip
  const int r0 = ty * 4;       // first local output row (LDS row = local_y + 1)
  float am[6], az[6], ap[6];
  #pragma unroll
  for (int k = 0; k < 6; ++k) am[k] = sS[r0 * LDSW + cb + k]       - sH[r0 * LDSW + cb + k];
  #pragma unroll
  for (int k = 0; k < 6; ++k) az[k] = sS[(r0 + 1) * LDSW + cb + k] - sH[(r0 + 1) * LDSW + cb + k];

  float acc = 0.0f;
  #pragma unroll
  for (int rr = 0; rr < 4; ++rr) {
    const int lr = r0 + 2 + rr;
    #pragma unroll
    for (int k = 0; k < 6; ++k) ap[k] = sS[lr * LDSW + cb + k] - sH[lr * LDSW + cb + k];
    #pragma unroll
    for (int j = 0; j < 4; ++j) {
      float gx = (am[j] - am[j + 2]) + 2.0f * (az[j] - az[j + 2]) + (ap[j] - ap[j + 2]);
      float gy = (am[j] + 2.0f * am[j + 1] + am[j + 2])
               - (ap[j] + 2.0f * ap[j + 1] + ap[j + 2]);
      acc += fabsf(gx) + fabsf(gy);
    }
    #pragma unroll
    for (int k = 0; k < 6; ++k) { am[k] = az[k]; az[k] = ap[k]; }
  }

  // ---- Block reduction: 256 -> 32 in LDS, 32 -> 1 via WMMA finisher ----
  red[tid] = acc;
  __syncthreads();
  if (tid < 32) {                          // uniform per-wave: wave0 EXEC all-ones
    float sum = 0.0f;
    #pragma unroll
    for (int k = 0; k < 8; ++k) sum += red[tid + 32 * k];
    float tot = wave_sum_wmma(sum);
    if (tid == 0)
      partial[(blockIdx.z * gridDim.y + blockIdx.y) * gridDim.x + blockIdx.x] = tot;
  }
}

__global__ __launch_bounds__(256)
void edge_sobel_finalize(const float* __restrict__ partial, float* __restrict__ out) {
  __shared__ float red[256];
  const int tid = threadIdx.x;
  double s = 0.0;
  for (int i = tid; i < NBLK; i += 256) s += (double)partial[i];
  red[tid] = (float)s;
  __syncthreads();
  if (tid < 32) {
    float sum = 0.0f;
    #pragma unroll
    for (int k = 0; k < 8; ++k) sum += red[tid + 32 * k];
    float tot = wave_sum_wmma(sum);
    if (tid == 0) out[0] = tot * (1.0f / 33554432.0f); // mean over 32*1024*1024
  }
}

extern "C" void kernel_launch(void* const* d_in, const int* in_sizes, int n_in,
                              void* d_out, int out_size, void* d_ws, size_t ws_size,
                              hipStream_t stream) {
  (void)in_sizes; (void)n_in; (void)out_size; (void)ws_size;
  const float* sr = (const float*)d_in[0];
  const float* hr = (const float*)d_in[1];
  float* partial  = (float*)d_ws;     // NBLK floats = 32 KB, fully overwritten
  float* out      = (float*)d_out;

  dim3 grid(IMG_W / TW, IMG_H / TH, N_IMG); // (8, 32, 32) = 8192 blocks
  edge_sobel_tile<<<grid, 256, 0, stream>>>(sr, hr, partial);
  edge_sobel_finalize<<<1, 256, 0, stream>>>(partial, out);
}